// ChamferDistance_35115652612616
// MI455X (gfx1250) — compile-verified
//
#include <hip/hip_runtime.h>
#include <hip/hip_bf16.h>
#include <math.h>

typedef __attribute__((ext_vector_type(2))) float v2f;
typedef __attribute__((ext_vector_type(8))) float v8f;

#define MAX_PTS 4096   // points of the opposing cloud staged in LDS (64 KB as float4)

// For each point of cloud A (per batch), find min Euclidean distance to all
// points of cloud B. Writes sqrt'ed min per A-point into outMin[b*Npts + n].
// grid = (ceil(Npts/128), B), block = 256 (8 waves, 16 A-rows per wave).
__global__ __launch_bounds__(256) void chamfer_nn_kernel(
    const float* __restrict__ A, const float* __restrict__ B,
    float* __restrict__ outMin, int Npts, int Mpts)
{
    __shared__ float4 sB[MAX_PTS];

    const int b = blockIdx.y;
    const float* __restrict__ Ab = A + (size_t)b * Npts * 3;
    const float* __restrict__ Bb = B + (size_t)b * Mpts * 3;

    // Stage opposing cloud into LDS as (x, y, z, ||p||^2)
    for (int i = threadIdx.x; i < Mpts; i += blockDim.x) {
        float x = Bb[3 * i + 0];
        float y = Bb[3 * i + 1];
        float z = Bb[3 * i + 2];
        sB[i] = make_float4(x, y, z, x * x + y * y + z * z);
    }
    __syncthreads();

    const int wave = threadIdx.x >> 5;
    const int lane = threadIdx.x & 31;
    const int half = lane >> 4;   // 0: K=0,1 lanes; 1: K=2,3 lanes
    const int li   = lane & 15;
    const int n0   = blockIdx.x * 128 + wave * 16;
    if (n0 >= Npts) return;       // whole-wave uniform skip (tail blocks)

    // A operand, 16x4 f32 layout: lanes 0-15 hold (K0,K1), lanes 16-31 (K2,K3)
    // for row M = lane%16. Row vector = (-2ax, -2ay, -2az, 1).
    {
    }
    const int   rowA = n0 + li;
    const float ax = Ab[3 * rowA + 0];
    const float ay = Ab[3 * rowA + 1];
    const float az = Ab[3 * rowA + 2];
    v2f aOp;
    aOp.x = half ? (-2.0f * az) : (-2.0f * ax);
    aOp.y = half ? 1.0f          : (-2.0f * ay);

    // Per-lane row norms: C/D vgpr i holds row (i + 8*half) of the tile.
    float rn[8];
#pragma unroll
    for (int i = 0; i < 8; ++i) {
        const int r = n0 + i + 8 * half;
        const float x = Ab[3 * r + 0];
        const float y = Ab[3 * r + 1];
        const float z = Ab[3 * r + 2];
        rn[i] = x * x + y * y + z * z;
    }

    float minv[8];
#pragma unroll
    for (int i = 0; i < 8; ++i) minv[i] = 3.0e38f;

    // Sweep all M points in 16-column tiles: D = A(16x4) * B(4x16) via WMMA.
    // B operand: lanes 0-15 read (bx,by) = float4.xy, lanes 16-31 read
    // (bz, ||b||^2) = float4.zw -> one aligned ds_load_b64 per lane per tile.
    for (int m0 = 0; m0 < Mpts; m0 += 16) {
        v2f bv = ((const v2f*)(&sB[m0 + li]))[half];
        v8f c  = {0.f, 0.f, 0.f, 0.f, 0.f, 0.f, 0.f, 0.f};
        v8f d  = __builtin_amdgcn_wmma_f32_16x16x4_f32(
            /*neg_a=*/false, aOp, /*neg_b=*/false, bv,
            /*c_mod=*/(short)0, c, /*reuse_a=*/false, /*reuse_b=*/false);
        // d[i] = ||b_col||^2 - 2 a_row . b_col ; add ||a_row||^2, track min.
#pragma unroll
        for (int i = 0; i < 8; ++i) {
            minv[i] = fminf(minv[i], d[i] + rn[i]);
        }
    }

    // Row-min across the 16 lanes of each half (cols live across lanes).
#pragma unroll
    for (int i = 0; i < 8; ++i) {
#pragma unroll
        for (int off = 1; off < 16; off <<= 1) {
            minv[i] = fminf(minv[i], __shfl_xor(minv[i], off, 32));
        }
    }

    if (li == 0) {
#pragma unroll
        for (int i = 0; i < 8; ++i) {
            const int r = n0 + i + 8 * half;
            outMin[(size_t)b * Npts + r] = sqrtf(fmaxf(minv[i], 0.0f));
        }
    }
}

// Single-block final reduction: out = sa * sum(a) + sb * sum(b).
__global__ __launch_bounds__(256) void chamfer_reduce_kernel(
    const float* __restrict__ a, int na, float sa,
    const float* __restrict__ b, int nb, float sb,
    float* __restrict__ out)
{
    float s1 = 0.0f, s2 = 0.0f;
    for (int i = threadIdx.x; i < na; i += blockDim.x) s1 += a[i];
    for (int i = threadIdx.x; i < nb; i += blockDim.x) s2 += b[i];
    float s = s1 * sa + s2 * sb;

#pragma unroll
    for (int off = 16; off > 0; off >>= 1) {
        s += __shfl_down(s, off, 32);
    }

    __shared__ float tmp[8];
    if ((threadIdx.x & 31) == 0) tmp[threadIdx.x >> 5] = s;
    __syncthreads();
    if (threadIdx.x == 0) {
        float t = 0.0f;
#pragma unroll
        for (int w = 0; w < 8; ++w) t += tmp[w];
        out[0] = t;
    }
}

extern "C" void kernel_launch(void* const* d_in, const int* in_sizes, int n_in,
                              void* d_out, int out_size, void* d_ws, size_t ws_size,
                              hipStream_t stream) {
    const float* in1 = (const float*)d_in[0];   // (B, N, 3)
    const float* in2 = (const float*)d_in[1];   // (B, M, 3)
    const int Bn = 8, K = 3;
    const int N = in_sizes[0] / (Bn * K);
    const int M = in_sizes[1] / (Bn * K);

    float* minA = (float*)d_ws;                 // B*N: dist1 (nearest in input2)
    float* minB = minA + (size_t)Bn * N;        // B*M: dist0 (nearest in input1)

    // dist1[b,n] = min_m ||in1 - in2||
    chamfer_nn_kernel<<<dim3((N + 127) / 128, Bn), dim3(256), 0, stream>>>(
        in1, in2, minA, N, M);
    // dist0[b,m] = min_n ||in1 - in2||  (same kernel, swapped roles)
    chamfer_nn_kernel<<<dim3((M + 127) / 128, Bn), dim3(256), 0, stream>>>(
        in2, in1, minB, M, N);

    // mean over b of (mean_m dist0 + mean_n dist1)
    chamfer_reduce_kernel<<<1, 256, 0, stream>>>(
        minA, Bn * N, 1.0f / (float)(Bn * N),
        minB, Bn * M, 1.0f / (float)(Bn * M),
        (float*)d_out);
}